// SpatialGraphConv_26336739459578
// MI455X (gfx1250) — compile-verified
//
#include <hip/hip_runtime.h>
#include <hip/hip_bf16.h>

// ---------------------------------------------------------------------------
// ChebConv (K=2) on MI455X / gfx1250.
//
// Dominant cost: dense contraction  Y_i = sum_{j,k} Tx_k[:,j,:] @ W[i,j,k]
//   => two GEMMs of M=N=2048, Ktot=8192  ~ 137 GFLOP fp32.
// Memory traffic ~235 MB => ~10us at 23.3 TB/s, so compute-bound on the
// fp32 matrix pipe: V_WMMA_F32_16X16X4_F32 (exact fp32, matches reference).
// Sparse propagate done with global fp32 atomics (2.7e8 adds, secondary cost).
// ---------------------------------------------------------------------------

#define NN   2048          // nodes == feature dim F
#define CC   2             // channels
#define KK   2             // Chebyshev order
#define EE   65536         // edges
#define FF   2048          // feature dim (== NN)
#define CF   (CC * FF)     // 4096 floats per node row

typedef __attribute__((ext_vector_type(2))) float v2f;
typedef __attribute__((ext_vector_type(8))) float v8f;

static const size_t TX1_COUNT = (size_t)NN * CF;   // 8,388,608 floats (33.5 MB)
// workspace layout (floats): [0, TX1) = Tx1 | [TX1, TX1+E) = wnorm | then deg[N]

// ---- zero Tx1 + deg ----
__global__ void zero_ws_kernel(float* ws) {
    size_t idx = (size_t)blockIdx.x * blockDim.x + threadIdx.x;
    if (idx < TX1_COUNT) {
        ws[idx] = 0.0f;                         // Tx1
    } else {
        size_t d = idx - TX1_COUNT;
        if (d < NN) ws[TX1_COUNT + EE + d] = 0.0f;   // deg (skip wnorm region)
    }
}

// ---- deg[row] += ew ----
__global__ void deg_kernel(const int* __restrict__ ei,
                           const float* __restrict__ ew,
                           float* __restrict__ deg) {
    int e = blockIdx.x * blockDim.x + threadIdx.x;
    if (e < EE) atomicAdd(&deg[ei[e]], ew[e]);
}

// ---- w_e = -rsqrt(deg[row]) * ew * rsqrt(deg[col]) ----
__global__ void norm_kernel(const int* __restrict__ ei,
                            const float* __restrict__ ew,
                            const float* __restrict__ deg,
                            float* __restrict__ wn) {
    int e = blockIdx.x * blockDim.x + threadIdx.x;
    if (e >= EE) return;
    int r = ei[e], c = ei[EE + e];
    float dr = deg[r], dc = deg[c];
    float ir = dr > 0.0f ? rsqrtf(dr) : 0.0f;
    float ic = dc > 0.0f ? rsqrtf(dc) : 0.0f;
    wn[e] = -ir * ew[e] * ic;
}

// ---- Tx1[col, :, :] += w_e * x[row, :, :]  (one block per edge) ----
__global__ __launch_bounds__(256) void scatter_kernel(const float* __restrict__ x,
                                                      const int* __restrict__ ei,
                                                      const float* __restrict__ wn,
                                                      float* __restrict__ tx1) {
    int e = blockIdx.x;
    float w = wn[e];
    if (w == 0.0f) return;
    int src = ei[e];
    int dst = ei[EE + e];
    const float* xs = x + (size_t)src * CF;
    float* td = tx1 + (size_t)dst * CF;
#pragma unroll
    for (int u = 0; u < CF / 256; ++u) {
        int idx = threadIdx.x + u * 256;
        atomicAdd(&td[idx], w * xs[idx]);
    }
}

// ---------------------------------------------------------------------------
// GEMM: out[n, i, g] = relu( sum_{j,k,f} A_{k,j}[n,f] * W[i,j,k][f,g] + bs_i[g] )
//   A_{0,j}[n,f] = x[n,j,f],  A_{1,j}[n,f] = Tx1[n,j,f]
// Block: 256 threads (8 waves), 128x128 output tile, wave = 32x64 sub-tile.
// LDS: A tile 128x32, B^T tile 128x32 (stride 36 floats: 16B-aligned float4
// stores, 8B-aligned v2f fragment reads).
// ---------------------------------------------------------------------------
__global__ __launch_bounds__(256) void cheb_gemm_kernel(const float* __restrict__ x,
                                                        const float* __restrict__ tx1,
                                                        const float* __restrict__ Wm,
                                                        const float* __restrict__ bias,
                                                        float* __restrict__ out) {
    __shared__ __align__(16) float As[128][36];   // [m][k]
    __shared__ __align__(16) float Bs[128][36];   // [g][k]  (B transposed)

    const int i  = blockIdx.z;            // output channel
    const int m0 = blockIdx.y * 128;
    const int n0 = blockIdx.x * 128;

    const int t    = threadIdx.x;
    const int lane = t & 31;
    const int wv   = t >> 5;              // wave id 0..7
    const int wm   = wv >> 1;             // 0..3  -> row offset wm*32
    const int wn   = wv & 1;              // 0..1  -> col offset wn*64
    const int l15  = lane & 15;
    const int half = lane >> 4;
    const int kb0  = half * 2;            // K base within 4-step per ISA layout

    v8f acc[2][4] = {};

    for (int p = 0; p < CC * KK; ++p) {
        const int j = p >> 1;             // input channel
        const int k = p & 1;              // Chebyshev order
        const float* Ab = (k == 0 ? x : tx1) + (size_t)j * FF;           // row stride CF
        const float* Bb = Wm + ((size_t)((i * CC + j) * KK + k)) * FF * FF;

        for (int fk = 0; fk < FF; fk += 32) {
            __syncthreads();
            // A tile: 128 rows x 32 k  (float4 per thread x4)
#pragma unroll
            for (int it = 0; it < 4; ++it) {
                int q  = t + it * 256;            // 0..1023
                int r  = q >> 3;                  // row 0..127
                int cv = (q & 7) << 2;            // k offset 0..28
                *(float4*)&As[r][cv] =
                    *(const float4*)(Ab + (size_t)(m0 + r) * CF + fk + cv);
            }
            // B tile: 32 k x 128 g, stored transposed
#pragma unroll
            for (int it = 0; it < 4; ++it) {
                int q  = t + it * 256;
                int f  = q >> 5;                  // k 0..31
                int gv = (q & 31) << 2;           // g offset 0..124
                float4 v = *(const float4*)(Bb + (size_t)(fk + f) * FF + n0 + gv);
                Bs[gv + 0][f] = v.x;
                Bs[gv + 1][f] = v.y;
                Bs[gv + 2][f] = v.z;
                Bs[gv + 3][f] = v.w;
            }
            // prefetch next K tile (global_prefetch_b8)
            if (fk + 32 < FF) {
                __builtin_prefetch(Ab + (size_t)(m0 + (t >> 1)) * CF + fk + 32, 0, 1);
                __builtin_prefetch(Bb + (size_t)(fk + 32 + (t & 31)) * FF + n0, 0, 1);
            }
            __syncthreads();

#pragma unroll
            for (int s = 0; s < 8; ++s) {
                const int kb = s * 4 + kb0;
                v2f a[2], bf[4];
                a[0] = *(const v2f*)&As[wm * 32 + l15][kb];
                a[1] = *(const v2f*)&As[wm * 32 + 16 + l15][kb];
#pragma unroll
                for (int nt = 0; nt < 4; ++nt)
                    bf[nt] = *(const v2f*)&Bs[wn * 64 + nt * 16 + l15][kb];
#pragma unroll
                for (int mt = 0; mt < 2; ++mt)
#pragma unroll
                    for (int nt = 0; nt < 4; ++nt)
                        acc[mt][nt] = __builtin_amdgcn_wmma_f32_16x16x4_f32(
                            false, a[mt], false, bf[nt],
                            (short)0, acc[mt][nt], false, false);
            }
        }
    }

    // epilogue: bias (summed over j) + relu, scatter per C/D VGPR layout
#pragma unroll
    for (int mt = 0; mt < 2; ++mt) {
#pragma unroll
        for (int nt = 0; nt < 4; ++nt) {
            int g = n0 + wn * 64 + nt * 16 + l15;
            float bs = bias[(size_t)(i * CC + 0) * FF + g]
                     + bias[(size_t)(i * CC + 1) * FF + g];
#pragma unroll
            for (int r = 0; r < 8; ++r) {
                int m = m0 + wm * 32 + mt * 16 + half * 8 + r;
                float v = acc[mt][nt][r] + bs;
                out[(size_t)m * CF + (size_t)i * FF + g] = fmaxf(v, 0.0f);
            }
        }
    }
}

extern "C" void kernel_launch(void* const* d_in, const int* in_sizes, int n_in,
                              void* d_out, int out_size, void* d_ws, size_t ws_size,
                              hipStream_t stream) {
    const float* x  = (const float*)d_in[0];   // (N, C, F)
    const int*   ei = (const int*)d_in[1];     // (2, E)
    const float* ew = (const float*)d_in[2];   // (E,)
    const float* Wm = (const float*)d_in[3];   // (C, C, K, F, F)
    const float* b  = (const float*)d_in[4];   // (C, C, F)
    float* out = (float*)d_out;                // (1, N, C, F)

    float* ws   = (float*)d_ws;
    float* tx1  = ws;                          // N*C*F floats
    float* wn   = ws + TX1_COUNT;              // E floats
    float* deg  = ws + TX1_COUNT + EE;         // N floats

    // 1) zero Tx1 + deg (every call; harness does not re-poison between replays)
    {
        size_t total = TX1_COUNT + NN;
        int blocks = (int)((total + 255) / 256);
        zero_ws_kernel<<<blocks, 256, 0, stream>>>(ws);
    }
    // 2) degree
    deg_kernel<<<EE / 256, 256, 0, stream>>>(ei, ew, deg);
    // 3) symmetric normalization
    norm_kernel<<<EE / 256, 256, 0, stream>>>(ei, ew, deg, wn);
    // 4) sparse propagate: Tx1 = L_hat @ x
    scatter_kernel<<<EE, 256, 0, stream>>>(x, ei, wn, tx1);
    // 5) dense WMMA contraction + bias + relu
    cheb_gemm_kernel<<<dim3(FF / 128, NN / 128, CC), 256, 0, stream>>>(
        x, tx1, Wm, b, out);
}